// Idefics3VisionEmbeddings_7301444403816
// MI455X (gfx1250) — compile-verified
//
#include <hip/hip_runtime.h>

typedef __attribute__((ext_vector_type(2))) float v2f;
typedef __attribute__((ext_vector_type(8))) float v8f;

#define D_DIM  1152
#define K_DIM  588
#define P_PER  676        // 26*26 patches per image
#define NPSIDE 26
#define HW     364
#define HWSQ   (364*364)
#define CHW    (3*364*364)
#define M_TOT  (32*676)   // 21632 rows = 169*128
#define BM     128
#define BN     128
#define BK     28         // 588 = 21*28 ; 196 = 7*28 (channel aligned)
#define NCHUNK (K_DIM / BK)   // 21

// ---------------------------------------------------------------------------
// Kernel 1: per-image NaViT bucketized position ids -> d_ws (int[21632])
// ---------------------------------------------------------------------------
__global__ void pos_ids_kernel(const unsigned char* __restrict__ mask,
                               int* __restrict__ pos_ids) {
  const int b = blockIdx.x;      // image
  const int t = threadIdx.x;     // 0..675 patch
  __shared__ int s_nbh, s_nbw;
  if (t == 0) {
    int nh = 0, nw = 0;
    for (int i = 0; i < NPSIDE; ++i) {
      nh += (mask[b * P_PER + i * NPSIDE] != 0);  // column 0
      nw += (mask[b * P_PER + i] != 0);           // row 0
    }
    s_nbh = nh; s_nbw = nw;
  }
  __syncthreads();
  const int h = t / NPSIDE, w = t % NPSIDE;
  const float eps = 1.0f - 1e-6f;
  const float fh = (float)h / (float)s_nbh * eps;
  const float fw = (float)w / (float)s_nbw * eps;
  int bh = 0, bw = 0;
  for (int i = 1; i < NPSIDE; ++i) {       // searchsorted(side='right')
    const float bd = (float)i / 26.0f;
    bh += (bd <= fh);
    bw += (bd <= fw);
  }
  const int pos = bh * NPSIDE + bw;
  pos_ids[b * P_PER + t] = mask[b * P_PER + t] ? pos : 0;
}

// ---------------------------------------------------------------------------
// Kernel 2: implicit-GEMM patch embedding, FP32 WMMA 16x16x4
//   out[row, d] = sum_k patch(row, k) * W[d, k] + bias[d] + pos_emb[pid[row], d]
// Workgroup tile 128x128 (8 waves, 4M x 2N); wave tile 32x64 (2x4 WMMA tiles).
// Double-buffered LDS: chunk k+1 global loads overlap chunk k WMMAs;
// one barrier per K-chunk.
// ---------------------------------------------------------------------------
__global__ __launch_bounds__(256)
void patch_embed_gemm(const float* __restrict__ px,
                      const float* __restrict__ W,
                      const float* __restrict__ bias,
                      const float* __restrict__ pos_emb,
                      const int*   __restrict__ pos_ids,
                      float* __restrict__ out) {
  __shared__ v2f A_lds[2][BM][15];        // k-pairs, row padded 14->15 pairs
  __shared__ v2f B_lds[2][BK / 2][BN];    // k-pairs: [k>>1][n] = {W[n][2k'],W[n][2k'+1]}

  const int tid   = threadIdx.x;
  const int wave  = tid >> 5;
  const int lane  = tid & 31;
  const int m_blk = blockIdx.x;        // 0..168
  const int n_blk = blockIdx.y;        // 0..8
  const int wm    = wave & 3;          // 4 M-subtiles of 32
  const int wn    = wave >> 2;         // 2 N-subtiles of 64

  // --- A-tile mapping: 128 rows x 28 k ; 2 threads/row, 14 contiguous k each
  const int a_row  = tid & 127;
  const int a_half = tid >> 7;               // 0/1 -> patch row kh within chunk
  const int g_row  = m_blk * BM + a_row;
  const int b_img  = g_row / P_PER;
  const int p      = g_row - b_img * P_PER;
  const int ph     = p / NPSIDE;
  const int pw     = p - ph * NPSIDE;
  const long a_base = (long)b_img * CHW + (long)(ph * 14 + a_half) * HW + (pw * 14);

  // --- B-tile mapping: 128 d x 28 k ; 2 threads/d, 14 contiguous k each
  const int b_d    = tid & 127;
  const int b_koff = (tid >> 7) * 14;
  const float* wrow = W + (long)(n_blk * BN + b_d) * K_DIM + b_koff;

  v2f a_stage[7], b_stage[7];

  auto load_global = [&](int kc) {
    const int k0  = kc * BK;
    const int c   = k0 / 196;                      // channel (chunk-uniform)
    const int kh0 = (k0 - c * 196) / 14;           // first patch row of chunk
    const float* asrc = px + a_base + (long)c * HWSQ + (long)kh0 * HW;
    #pragma unroll
    for (int j = 0; j < 7; ++j) {
      a_stage[j].x = asrc[2 * j];
      a_stage[j].y = asrc[2 * j + 1];
    }
    const float* wsrc = wrow + k0;
    #pragma unroll
    for (int j = 0; j < 7; ++j) {
      b_stage[j].x = wsrc[2 * j];
      b_stage[j].y = wsrc[2 * j + 1];
    }
  };
  auto store_lds = [&](int buf) {
    #pragma unroll
    for (int j = 0; j < 7; ++j)
      A_lds[buf][a_row][a_half * 7 + j] = a_stage[j];
    #pragma unroll
    for (int j = 0; j < 7; ++j)
      B_lds[buf][(b_koff >> 1) + j][b_d] = b_stage[j];
  };

  v8f acc[2][4] = {};

  // prologue: fill buffer 0
  load_global(0);
  store_lds(0);
  __syncthreads();

  for (int kc = 0; kc < NCHUNK; ++kc) {
    const int cur = kc & 1;
    if (kc + 1 < NCHUNK)
      load_global(kc + 1);                         // overlap with WMMAs below

    #pragma unroll
    for (int ks = 0; ks < 7; ++ks) {               // 7 x K=4
      const int kp = ks * 2 + (lane >> 4);         // lane's k-pair index
      v2f a[2], bfrag[4];
      #pragma unroll
      for (int mt = 0; mt < 2; ++mt)
        a[mt] = A_lds[cur][wm * 32 + mt * 16 + (lane & 15)][kp];
      #pragma unroll
      for (int nt = 0; nt < 4; ++nt)
        bfrag[nt] = B_lds[cur][kp][wn * 64 + nt * 16 + (lane & 15)];
      #pragma unroll
      for (int mt = 0; mt < 2; ++mt)
        #pragma unroll
        for (int nt = 0; nt < 4; ++nt)
          acc[mt][nt] = __builtin_amdgcn_wmma_f32_16x16x4_f32(
              false, a[mt], false, bfrag[nt], (short)0, acc[mt][nt],
              false, false);
    }

    if (kc + 1 < NCHUNK)
      store_lds(cur ^ 1);        // other buffer: safe, its readers passed the
                                 // previous barrier already
    __syncthreads();
  }

  // --- epilogue: + bias + pos_emb gather, FP32 store ---
  const int row_base = m_blk * BM + wm * 32;
  const int col_base = n_blk * BN + wn * 64;
  float bc[4];
  #pragma unroll
  for (int nt = 0; nt < 4; ++nt)
    bc[nt] = bias[col_base + nt * 16 + (lane & 15)];

  #pragma unroll
  for (int mt = 0; mt < 2; ++mt) {
    #pragma unroll
    for (int v = 0; v < 8; ++v) {
      const int row = row_base + mt * 16 + (lane >> 4) * 8 + v;
      const int pid = pos_ids[row];
      const float* pe = pos_emb + (long)pid * D_DIM;
      float* op = out + (long)row * D_DIM;
      #pragma unroll
      for (int nt = 0; nt < 4; ++nt) {
        const int col = col_base + nt * 16 + (lane & 15);
        op[col] = acc[mt][nt][v] + bc[nt] + pe[col];
      }
    }
  }
}

// ---------------------------------------------------------------------------
extern "C" void kernel_launch(void* const* d_in, const int* in_sizes, int n_in,
                              void* d_out, int out_size, void* d_ws, size_t ws_size,
                              hipStream_t stream) {
  const float*         px   = (const float*)d_in[0];
  const unsigned char* mask = (const unsigned char*)d_in[1];  // bool mask
  const float*         W    = (const float*)d_in[2];
  const float*         bias = (const float*)d_in[3];
  const float*         pose = (const float*)d_in[4];
  float* out    = (float*)d_out;
  int* pos_ids  = (int*)d_ws;                 // 21632 ints

  pos_ids_kernel<<<32, P_PER, 0, stream>>>(mask, pos_ids);

  dim3 grid(M_TOT / BM, D_DIM / BN);          // (169, 9)
  patch_embed_gemm<<<grid, 256, 0, stream>>>(px, W, bias, pose, pos_ids, out);
}